// GNN_64201171140658
// MI455X (gfx1250) — compile-verified
//
#include <hip/hip_runtime.h>

// GNN GeneralConv x2, refactored via linearity of segment_sum:
//   segment_sum(x[src] @ W, dst) == segment_sum(x[src], dst) @ W
// Edge phase = scatter-add of raw features (memory-bound, ~440 MB HBM;
//   edge_attr streamed with TH=NT so the L2-resident gather/scatter working
//   set -- x, h, accumulators, ~77 MB << 192 MB L2 -- is not evicted).
// Node phase = [N,64]@[64,64] matmuls with V_WMMA_F32_16X16X4_F32; weight
//   B-fragments staged pair-interleaved in LDS -> single ds_load_b64 each.

#define N_NODES 100000
#define N_EDGES 1600000
#define D 64

typedef float v2f __attribute__((ext_vector_type(2)));
typedef float v4f __attribute__((ext_vector_type(4)));
typedef float v8f __attribute__((ext_vector_type(8)));

// ---------------------------------------------------------------- zero fill
__global__ __launch_bounds__(256) void GNN_zero_f4(v4f* __restrict__ p,
                                                   unsigned n4) {
  unsigned i = blockIdx.x * 256u + threadIdx.x;
  if (i < n4) p[i] = (v4f){0.f, 0.f, 0.f, 0.f};
}

// --------------------------------------------- edge pass 1: x, edge_attr, deg
// 16 threads per edge, 4 consecutive floats each (float4 loads, f32 atomics).
// edge_attr is single-use: non-temporal load keeps it out of L2.
__global__ __launch_bounds__(256) void GNN_edge_pass1(
    const float* __restrict__ x, const float* __restrict__ ea,
    const int* __restrict__ ei, float* __restrict__ aggx,
    float* __restrict__ agge, float* __restrict__ deg) {
  const unsigned tid = blockIdx.x * 256u + threadIdx.x;
  const unsigned e = tid >> 4;
  const unsigned f = (tid & 15u) << 2;
  if (e >= N_EDGES) return;
  const int src = ei[e];
  const int dst = ei[N_EDGES + e];
  const v4f xv = *(const v4f*)(x + (size_t)src * D + f);  // reused: keep in L2
  const v4f av =
      __builtin_nontemporal_load((const v4f*)(ea + (size_t)e * D + f));
  float* px = aggx + (size_t)dst * D + f;
  float* pe = agge + (size_t)dst * D + f;
  atomicAdd(px + 0, xv.x); atomicAdd(px + 1, xv.y);
  atomicAdd(px + 2, xv.z); atomicAdd(px + 3, xv.w);
  atomicAdd(pe + 0, av.x); atomicAdd(pe + 1, av.y);
  atomicAdd(pe + 2, av.z); atomicAdd(pe + 3, av.w);
  if ((tid & 15u) == 0u) atomicAdd(deg + dst, 1.0f);
}

// --------------------------------------------------- edge pass 2: h[src] only
__global__ __launch_bounds__(256) void GNN_edge_pass2(
    const float* __restrict__ h, const int* __restrict__ ei,
    float* __restrict__ aggh) {
  const unsigned tid = blockIdx.x * 256u + threadIdx.x;
  const unsigned e = tid >> 4;
  const unsigned f = (tid & 15u) << 2;
  if (e >= N_EDGES) return;
  const int src = ei[e];
  const int dst = ei[N_EDGES + e];
  const v4f hv = *(const v4f*)(h + (size_t)src * D + f);  // reused: keep in L2
  float* ph = aggh + (size_t)dst * D + f;
  atomicAdd(ph + 0, hv.x); atomicAdd(ph + 1, hv.y);
  atomicAdd(ph + 2, hv.z); atomicAdd(ph + 3, hv.w);
}

// ------------------------------------------------- node update (WMMA f32)
// out = aggm@Wm + agge@We + xs@Ws + deg*(bm+be) + bs  [+ relu / NT store]
// One wave per 16-node tile; 4 accumulators cover the 64 output features.
// A-frag (16x4 f32): lanes 0-15 rows M=0..15, K={0,1}; lanes 16-31 K={2,3}.
// B-frag (4x16 f32): VGPR0 = K {0|2}, VGPR1 = K {1|3}, N = lane&15.
// C/D (16x16 f32): VGPR r -> M=r (lanes 0-15) / M=r+8 (lanes 16-31), N=lane&15.
// Weights staged in LDS pair-interleaved: ldsW[m][p*64+c] = {W[2p][c],W[2p+1][c]}
// so each B fragment is a single ds_load_b64.
__global__ __launch_bounds__(256) void GNN_node_update(
    const float* __restrict__ aggm, const float* __restrict__ agge,
    const float* __restrict__ xs, const float* __restrict__ Wm,
    const float* __restrict__ We, const float* __restrict__ Ws,
    const float* __restrict__ bm, const float* __restrict__ be,
    const float* __restrict__ bs, const float* __restrict__ deg,
    float* __restrict__ out, int do_relu) {
  __shared__ v2f ldsW[3 * 2048];  // 48 KB of 320 KB WGP LDS

  // ---- stage Wm|We|Ws into LDS, pair-interleaved (all threads participate)
  const float* Wsrc[3] = {Wm, We, Ws};
#pragma unroll
  for (int m = 0; m < 3; ++m) {
    const float* W = Wsrc[m];
    for (int idx = threadIdx.x; idx < 2048; idx += 256) {
      const int p = idx >> 6;   // K pair index 0..31
      const int c = idx & 63;   // output column
      v2f v;
      v.x = W[(size_t)(2 * p) * D + c];
      v.y = W[(size_t)(2 * p + 1) * D + c];
      ldsW[m * 2048 + idx] = v;
    }
  }
  __syncthreads();

  const int lane = threadIdx.x & 31;
  const int wave = threadIdx.x >> 5;
  const int tile = blockIdx.x * 8 + wave;
  const int row0 = tile * 16;
  if (row0 < N_NODES) {  // wave-uniform: EXEC stays all-ones for WMMA
    const int mrow = lane & 15;   // row of this lane's A-frag element
    const int khalf = lane >> 4;  // which K pair this half-wave holds
    const int col = lane & 15;    // N index within a 16-wide C/D tile

    v8f acc[4] = {};

    const float* aM_base = aggm + (size_t)(row0 + mrow) * D + khalf * 2;
    const float* aE_base = agge + (size_t)(row0 + mrow) * D + khalf * 2;
    const float* aS_base = xs + (size_t)(row0 + mrow) * D + khalf * 2;

    for (int kk = 0; kk < 16; ++kk) {
      const v2f aM = *(const v2f*)(aM_base + kk * 4);
      const v2f aE = *(const v2f*)(aE_base + kk * 4);
      const v2f aS = *(const v2f*)(aS_base + kk * 4);
      const int pairIdx = kk * 2 + khalf;  // K pair this half-wave needs
#pragma unroll
      for (int nt = 0; nt < 4; ++nt) {
        const int c = nt * 16 + col;
        const v2f bM = ldsW[0 * 2048 + pairIdx * 64 + c];
        const v2f bE = ldsW[1 * 2048 + pairIdx * 64 + c];
        const v2f bS = ldsW[2 * 2048 + pairIdx * 64 + c];
        acc[nt] = __builtin_amdgcn_wmma_f32_16x16x4_f32(
            false, aM, false, bM, (short)0, acc[nt], false, false);
        acc[nt] = __builtin_amdgcn_wmma_f32_16x16x4_f32(
            false, aE, false, bE, (short)0, acc[nt], false, false);
        acc[nt] = __builtin_amdgcn_wmma_f32_16x16x4_f32(
            false, aS, false, bS, (short)0, acc[nt], false, false);
      }
    }

    float degv[8];
#pragma unroll
    for (int r = 0; r < 8; ++r) degv[r] = deg[row0 + khalf * 8 + r];

#pragma unroll
    for (int nt = 0; nt < 4; ++nt) {
      const int c = nt * 16 + col;
      const float mb = bm[c] + be[c];
      const float bsv = bs[c];
#pragma unroll
      for (int r = 0; r < 8; ++r) {
        float v = acc[nt][r] + degv[r] * mb + bsv;
        float* dst = &out[(size_t)(row0 + khalf * 8 + r) * D + c];
        if (do_relu) {
          *dst = fmaxf(v, 0.0f);  // h: re-read by edge pass 2, keep cached
        } else {
          __builtin_nontemporal_store(v, dst);  // final output: single-use
        }
      }
    }
  }
}

extern "C" void kernel_launch(void* const* d_in, const int* in_sizes, int n_in,
                              void* d_out, int out_size, void* d_ws,
                              size_t ws_size, hipStream_t stream) {
  (void)in_sizes; (void)n_in; (void)out_size; (void)ws_size;
  // setup_inputs order: x, edge_index, edge_attr, Wm1,We1,Ws1,Wm2,We2,Ws2,
  //                     bm1,be1,bs1,bm2,be2,bs2
  const float* x = (const float*)d_in[0];
  const int* ei = (const int*)d_in[1];  // [2,E]: src = ei[0:E], dst = ei[E:2E]
  const float* ea = (const float*)d_in[2];
  const float* Wm1 = (const float*)d_in[3];
  const float* We1 = (const float*)d_in[4];
  const float* Ws1 = (const float*)d_in[5];
  const float* Wm2 = (const float*)d_in[6];
  const float* We2 = (const float*)d_in[7];
  const float* Ws2 = (const float*)d_in[8];
  const float* bm1 = (const float*)d_in[9];
  const float* be1 = (const float*)d_in[10];
  const float* bs1 = (const float*)d_in[11];
  const float* bm2 = (const float*)d_in[12];
  const float* be2 = (const float*)d_in[13];
  const float* bs2 = (const float*)d_in[14];

  // workspace layout: [agg (N*D) | agge (N*D) | deg (N) | h (N*D)]
  float* agg = (float*)d_ws;
  float* agge = agg + (size_t)N_NODES * D;
  float* deg = agge + (size_t)N_NODES * D;
  float* h = deg + N_NODES;

  const int edge_blocks = (N_EDGES * 16) / 256;    // exact: 100000
  const int node_blocks = (N_NODES / 16 + 7) / 8;  // 782 (tile guard)

  // 1. zero agg|agge|deg (contiguous region, count divisible by 4)
  {
    unsigned n4 = (2u * N_NODES * D + N_NODES) / 4u;
    GNN_zero_f4<<<(n4 + 255u) / 256u, 256, 0, stream>>>((v4f*)agg, n4);
  }
  // 2. edge pass 1: agg += x[src], agge += edge_attr, deg += 1 per edge
  GNN_edge_pass1<<<edge_blocks, 256, 0, stream>>>(x, ea, ei, agg, agge, deg);
  // 3. layer 1: h = relu(agg@Wm1 + agge@We1 + x@Ws1 + deg*(bm1+be1) + bs1)
  GNN_node_update<<<node_blocks, 256, 0, stream>>>(
      agg, agge, x, Wm1, We1, Ws1, bm1, be1, bs1, deg, h, 1);
  // 4. re-zero agg for the h-aggregation
  {
    unsigned n4 = (unsigned)((size_t)N_NODES * D / 4);
    GNN_zero_f4<<<(n4 + 255u) / 256u, 256, 0, stream>>>((v4f*)agg, n4);
  }
  // 5. edge pass 2: agg += h[src]
  GNN_edge_pass2<<<edge_blocks, 256, 0, stream>>>(h, ei, agg);
  // 6. layer 2: out = agg@Wm2 + agge@We2 + h@Ws2 + deg*(bm2+be2) + bs2
  GNN_node_update<<<node_blocks, 256, 0, stream>>>(
      agg, agge, h, Wm2, We2, Ws2, bm2, be2, bs2, deg, (float*)d_out, 0);
}